// S5Layer_59279138619403
// MI455X (gfx1250) — compile-verified
//
#include <hip/hip_runtime.h>

// ---------------------------------------------------------------------------
// S5 layer for MI455X (gfx1250, wave32):
//   1) prep:  Lambda_bar, B_bar -> W1 (K=H x N=2P), 2*C folded -> W2 (2P x H)
//   2) gemm:  Bu = x @ W1          (fp32 WMMA 16x16x4)
//   3) scan:  x_t = lam*x_{t-1}+Bu (chunked parallel scan, in-place)
//   4) gemm:  ys = xs @ W2 + D*x   (fp32 WMMA 16x16x4, fused epilogue)
// HBM-bound (~800 MB traffic -> ~35us at 23.3 TB/s); fp32 WMMA keeps full
// precision at zero roofline cost.
// ---------------------------------------------------------------------------

#define SB 16
#define SL 8192
#define SH 256     // K and N of both GEMMs (N = 2*P = 256 as well)
#define SP 128
#define BL (SB * SL)          // 131072 rows
#define LDSTR 260             // padded LDS row stride (floats): conflict-free, 16B-aligned rows

typedef float v2f __attribute__((ext_vector_type(2)));
typedef float v8f __attribute__((ext_vector_type(8)));

__device__ __forceinline__ float2 cmul(float2 a, float2 b) {
    return make_float2(a.x * b.x - a.y * b.y, a.x * b.y + a.y * b.x);
}
__device__ __forceinline__ float2 cadd(float2 a, float2 b) {
    return make_float2(a.x + b.x, a.y + b.y);
}

// ---------------------------------------------------------------------------
// Kernel 1: build the two 256x256 real weight matrices.
//   W1[h*256 + 2p]   = Re(B_bar[p][h]),  W1[h*256 + 2p+1] = Im(B_bar[p][h])
//   W2[(2p)*256 + h] = 2*C_re[h][p],     W2[(2p+1)*256+h] = -2*C_im[h][p]
// ---------------------------------------------------------------------------
__global__ __launch_bounds__(256) void prep_kernel(
    const float* __restrict__ Lre, const float* __restrict__ Lim,
    const float* __restrict__ Bre, const float* __restrict__ Bim,
    const float* __restrict__ Cre, const float* __restrict__ Cim,
    const float* __restrict__ lstep,
    float* __restrict__ W1, float* __restrict__ W2)
{
    int idx = blockIdx.x * blockDim.x + threadIdx.x;   // 0 .. P*H-1
    int p = idx >> 8;          // / 256
    int h = idx & 255;
    float dt = expf(lstep[p]);
    float lr = Lre[p], li = Lim[p];
    float er  = expf(lr * dt);
    float lbr = er * cosf(li * dt);
    float lbi = er * sinf(li * dt);
    // f = (Lambda_bar - 1) / Lambda
    float inv = 1.0f / (lr * lr + li * li);
    float nr = lbr - 1.0f, ni = lbi;
    float fr = (nr * lr + ni * li) * inv;
    float fi = (ni * lr - nr * li) * inv;
    float br = Bre[p * SH + h], bi = Bim[p * SH + h];
    W1[h * SH + 2 * p]     = fr * br - fi * bi;
    W1[h * SH + 2 * p + 1] = fr * bi + fi * br;
    W2[(2 * p) * SH + h]     =  2.0f * Cre[h * SP + p];
    W2[(2 * p + 1) * SH + h] = -2.0f * Cim[h * SP + p];
}

// ---------------------------------------------------------------------------
// Kernel 2/4: Out(BL x 256) = Xin(BL x 256) @ W(256 x 256) [+ D[col]*Xres]
// block = 256 threads (8 waves); block M-tile = 128 rows in LDS;
// each wave computes a 16-row strip; 4 N-tiles per accumulator group share
// one A fragment (V_WMMA_F32_16X16X4_F32). ADDRES is compile-time.
// ---------------------------------------------------------------------------
template <int ADDRES>
__global__ __launch_bounds__(256) void gemm256_wmma(
    const float* __restrict__ Xin, const float* __restrict__ W,
    float* __restrict__ Out,
    const float* __restrict__ Dvec, const float* __restrict__ Xres)
{
    __shared__ float tileX[128 * LDSTR];   // 128 rows x 256 K (padded)

    int tid = threadIdx.x;
    long rowBase = (long)blockIdx.x * 128;

    // cooperative tile load: 128*256 floats = 8192 float4 slots, 32 per thread
    #pragma unroll 4
    for (int i = 0; i < 32; ++i) {
        int slot = tid + i * 256;          // 0..8191
        int r  = slot >> 6;                // row 0..127
        int c4 = (slot & 63) << 2;         // col 0..252 step 4
        float4 v = *(const float4*)(Xin + (rowBase + r) * SH + c4);
        float* dst = &tileX[r * LDSTR + c4];
        dst[0] = v.x; dst[1] = v.y; dst[2] = v.z; dst[3] = v.w;
    }
    __syncthreads();

    int wave = tid >> 5;            // 0..7 -> rows wave*16 .. wave*16+15
    int lane = tid & 31;
    int l16  = lane & 15;
    int hi   = lane >> 4;           // 0: K pair {0,1} / 1: K pair {2,3}

    const float* arow = &tileX[(wave * 16 + l16) * LDSTR];

    for (int g = 0; g < 4; ++g) {          // 4 groups x 4 N-tiles = N=256
        int n0 = g * 64;
        const float* wc = W + n0 + l16;
        v8f acc0 = {}, acc1 = {}, acc2 = {}, acc3 = {};
        #pragma unroll 4
        for (int k0 = 0; k0 < SH; k0 += 4) {
            // A 16x4 fragment: lanes 0-15 hold K=k0,k0+1 ; lanes 16-31 K=k0+2,k0+3
            float2 av = *(const float2*)(arow + k0 + 2 * hi);
            v2f a; a.x = av.x; a.y = av.y;
            // B 4x16 fragments: v0 rows k0/k0+1 (half-waves), v1 rows k0+2/k0+3
            int r0 = (k0 + hi) * SH;
            int r1 = (k0 + 2 + hi) * SH;
            v2f b0; b0.x = wc[r0];      b0.y = wc[r1];
            v2f b1; b1.x = wc[r0 + 16]; b1.y = wc[r1 + 16];
            v2f b2; b2.x = wc[r0 + 32]; b2.y = wc[r1 + 32];
            v2f b3; b3.x = wc[r0 + 48]; b3.y = wc[r1 + 48];
            acc0 = __builtin_amdgcn_wmma_f32_16x16x4_f32(false, a, false, b0, (short)0, acc0, false, false);
            acc1 = __builtin_amdgcn_wmma_f32_16x16x4_f32(false, a, false, b1, (short)0, acc1, false, false);
            acc2 = __builtin_amdgcn_wmma_f32_16x16x4_f32(false, a, false, b2, (short)0, acc2, false, false);
            acc3 = __builtin_amdgcn_wmma_f32_16x16x4_f32(false, a, false, b3, (short)0, acc3, false, false);
        }
        // C/D layout: VGPR r -> (M=r, lanes 0-15) / (M=r+8, lanes 16-31)
        v8f accs[4] = {acc0, acc1, acc2, acc3};
        #pragma unroll
        for (int j = 0; j < 4; ++j) {
            int col = n0 + j * 16 + l16;
            float dj = ADDRES ? Dvec[col] : 0.0f;
            #pragma unroll
            for (int r = 0; r < 8; ++r) {
                long grow = rowBase + wave * 16 + r + 8 * hi;
                float v = accs[j][r];
                if (ADDRES) v += dj * Xres[grow * SH + col];
                Out[grow * SH + col] = v;
            }
        }
    }
}

// ---------------------------------------------------------------------------
// Kernel 3: in-place chunked parallel scan over L for one (b,p) per block.
//   x_t = lam * x_{t-1} + Bu_t ; lam = exp(Lambda[p] * dt)
// 256 threads x 32 contiguous elements each = 8192 = L.
// Segment carries combined with a Hillis-Steele scan (multiplier lam^(32*2^k)).
// ---------------------------------------------------------------------------
__global__ __launch_bounds__(256) void scan_kernel(
    float* __restrict__ Bu,
    const float* __restrict__ Lre, const float* __restrict__ Lim,
    const float* __restrict__ lstep)
{
    __shared__ float2 sc[2][256];

    int b = blockIdx.x >> 7;       // / P
    int p = blockIdx.x & 127;
    int tid = threadIdx.x;

    float dt = expf(lstep[p]);
    float lr = Lre[p] * dt, li = Lim[p] * dt;
    float er = expf(lr);
    float2 lam = make_float2(er * cosf(li), er * sinf(li));

    // complex element l lives at (float2*)Bu + (b*L + l)*128 + p
    float2* base = (float2*)Bu + (long)b * SL * SP + p;
    long l0 = (long)tid * 32;

    float2 v[32];
    #pragma unroll
    for (int i = 0; i < 32; ++i) v[i] = base[(l0 + i) * SP];

    // local segment scan (carry-in 0)
    float2 s = make_float2(0.f, 0.f);
    #pragma unroll
    for (int i = 0; i < 32; ++i) s = cadd(cmul(lam, s), v[i]);

    // lam^32 by repeated squaring
    float2 m = lam;
    #pragma unroll
    for (int i = 0; i < 5; ++i) m = cmul(m, m);

    sc[0][tid] = s;
    int buf = 0;
    for (int d = 1; d < 256; d <<= 1) {
        __syncthreads();
        float2 cur = sc[buf][tid];
        if (tid >= d) cur = cadd(cmul(m, sc[buf][tid - d]), cur);
        sc[buf ^ 1][tid] = cur;
        buf ^= 1;
        m = cmul(m, m);
    }
    __syncthreads();
    float2 carry = (tid == 0) ? make_float2(0.f, 0.f) : sc[buf][tid - 1];

    // replay with carry and write xs in place
    #pragma unroll
    for (int i = 0; i < 32; ++i) {
        carry = cadd(cmul(lam, carry), v[i]);
        base[(l0 + i) * SP] = carry;
    }
}

// ---------------------------------------------------------------------------
extern "C" void kernel_launch(void* const* d_in, const int* in_sizes, int n_in,
                              void* d_out, int out_size, void* d_ws, size_t ws_size,
                              hipStream_t stream) {
    const float* x    = (const float*)d_in[0];
    const float* Lre  = (const float*)d_in[1];
    const float* Lim  = (const float*)d_in[2];
    const float* Bre  = (const float*)d_in[3];
    const float* Bim  = (const float*)d_in[4];
    const float* Cre  = (const float*)d_in[5];
    const float* Cim  = (const float*)d_in[6];
    const float* Dv   = (const float*)d_in[7];
    const float* lstp = (const float*)d_in[8];

    float* ws = (float*)d_ws;
    float* W1 = ws;                       // 256*256
    float* W2 = ws + 65536;               // 256*256
    float* Bu = ws + 131072;              // BL*256 floats (complex interleaved), scanned in place

    prep_kernel<<<(SP * SH) / 256, 256, 0, stream>>>(Lre, Lim, Bre, Bim, Cre, Cim, lstp, W1, W2);
    gemm256_wmma<0><<<BL / 128, 256, 0, stream>>>(x, W1, Bu, nullptr, nullptr);
    scan_kernel<<<SB * SP, 256, 0, stream>>>(Bu, Lre, Lim, lstp);
    gemm256_wmma<1><<<BL / 128, 256, 0, stream>>>(Bu, W2, (float*)d_out, Dv, x);
}